// GATWithSkip_43052752175813
// MI455X (gfx1250) — compile-verified
//
#include <hip/hip_runtime.h>
#include <hip/hip_bf16.h>
#include <math.h>

#define N_NODES 50000
#define N_EDGES 800000
#define ETOT    850000
#define IN_CH   128
#define CC      64
#define HD4     4
#define HC      256
#define SLOPE   0.2f
#define EPS_BN  1e-5f
#define PADK    4        // K+4 == 4 (mod 64) for K=128/256 -> conflict-free ds_load_b64

typedef float v2f __attribute__((ext_vector_type(2)));
typedef float v8f __attribute__((ext_vector_type(8)));

static inline int cdiv(int a, int b) { return (a + b - 1) / b; }

__device__ __forceinline__ void atomicMaxF(float* addr, float val) {
  int iv = __float_as_int(val);
  if (iv >= 0) atomicMax((int*)addr, iv);
  else         atomicMin((unsigned int*)addr, (unsigned int)iv);
}

// ---------------- fill ----------------
__global__ void k_fill(float* __restrict__ p, float v, int n) {
  int t = blockIdx.x * blockDim.x + threadIdx.x;
  if (t < n) p[t] = v;
}

// ---------------- GEMM: C[M,Nn] = A[M,K] @ W[K,Nn] (+bias) ----------------
// fp32 WMMA 16x16x4.  blockDim = 256 (8 waves); each wave: 16 rows x 64 cols.
// Weight slice W[:, n0:n0+64] staged transposed in LDS (wsT[n][k], pitch K+4).
// A fragments double-buffered in registers: next 16-K batch issued before the
// current batch's 16 WMMAs so global latency overlaps matrix math.
// Frag layouts (lane l, vgpr v):
//   A 16x4:  row M = l%16,  K = k0 + (l>>4)*2 + v
//   B 4x16:  col N = l%16,  K = k0 + (l>>4)*2 + v
//   C/D:     M = (l>>4)*8 + g, N = l%16
__global__ void k_gemm_wmma(const float* __restrict__ A, const float* __restrict__ W,
                            const float* __restrict__ bias, float* __restrict__ C,
                            int Mt, int K, int Nn) {
  extern __shared__ float wsT[];                 // [64][K+PADK]
  const int Ks = K + PADK;
  const int tid = threadIdx.x;
  const int n0 = blockIdx.y * 64;

  // cooperative transposed staging of the 64-wide weight slice
  for (int idx = tid; idx < 64 * K; idx += 256) {
    int k = idx >> 6;                            // coalesced over n
    int n = idx & 63;
    wsT[n * Ks + k] = W[(size_t)k * Nn + n0 + n];
  }
  __syncthreads();

  int wv = tid >> 5, lane = tid & 31;
  int rt = blockIdx.x * 8 + wv;
  if (rt >= Mt) return;                          // wave-uniform tail exit
  int lm = lane & 15;
  int lk = (lane >> 4) << 1;
  const float* Arow = A + (size_t)(rt * 16 + lm) * K + lk;
  const float* bp   = wsT + lm * Ks + lk;        // 8B-aligned per-lane LDS base

  v8f acc[4];
#pragma unroll
  for (int t = 0; t < 4; ++t)
#pragma unroll
    for (int g = 0; g < 8; ++g) acc[t][g] = 0.0f;

  v2f a[4], an[4];
#pragma unroll
  for (int j = 0; j < 4; ++j) {                  // preload K batch 0..15
    a[j].x = Arow[j * 4];
    a[j].y = Arow[j * 4 + 1];
  }

  int k0 = 0;
  while (true) {
    int kn = k0 + 16;
    bool more = kn < K;
    if (more) {                                  // prefetch next A batch (one clause)
#pragma unroll
      for (int j = 0; j < 4; ++j) {
        an[j].x = Arow[kn + j * 4];
        an[j].y = Arow[kn + j * 4 + 1];
      }
    }
#pragma unroll
    for (int j = 0; j < 4; ++j) {
      int kk = k0 + j * 4;
#pragma unroll
      for (int t = 0; t < 4; ++t) {
        v2f b = *(const v2f*)(bp + t * 16 * Ks + kk);   // ds_load_b64
        acc[t] = __builtin_amdgcn_wmma_f32_16x16x4_f32(
            false, a[j], false, b, (short)0, acc[t], false, false);
      }
    }
    if (!more) break;
#pragma unroll
    for (int j = 0; j < 4; ++j) a[j] = an[j];
    k0 = kn;
  }

  int rbase = rt * 16 + ((lane >> 4) << 3);
#pragma unroll
  for (int t = 0; t < 4; ++t) {
    int col = n0 + t * 16 + lm;
    float bv = bias ? bias[col] : 0.0f;
#pragma unroll
    for (int g = 0; g < 8; ++g)
      C[(size_t)(rbase + g) * Nn + col] = acc[t][g] + bv;
  }
}

// --------- per-(node,head) attention dot products: al = sum_c h*a ---------
__global__ void k_attn_dots(const float* __restrict__ h, const float* __restrict__ a_s,
                            const float* __restrict__ a_d, float* __restrict__ als,
                            float* __restrict__ ald, int NH, int Hm) {
  int wid  = (blockIdx.x * blockDim.x + threadIdx.x) >> 5; // one wave per (node,head)
  int lane = threadIdx.x & 31;
  if (wid >= NH) return;
  int hd = wid & Hm;
  const float* hp  = h + (size_t)wid * CC;
  const float* asp = a_s + hd * CC;
  const float* adp = a_d + hd * CC;
  float ss = 0.f, sd = 0.f;
  for (int c = lane; c < CC; c += 32) {
    float v = hp[c];
    ss += v * asp[c];
    sd += v * adp[c];
  }
#pragma unroll
  for (int off = 16; off > 0; off >>= 1) {
    ss += __shfl_xor(ss, off, 32);
    sd += __shfl_xor(sd, off, 32);
  }
  if (lane == 0) { als[wid] = ss; ald[wid] = sd; }
}

// ---------------- edge pass 1: segment max ----------------
__global__ void k_edge_max(const long long* __restrict__ ei, int Hshift, int Hm,
                           const float* __restrict__ als, const float* __restrict__ ald,
                           float* __restrict__ m) {
  int t = blockIdx.x * blockDim.x + threadIdx.x;
  int total = ETOT << Hshift;
  if (t >= total) return;
  int e = t >> Hshift, hd = t & Hm, H = Hm + 1;
  int s, d;
  if (e < N_EDGES) { s = (int)ei[e]; d = (int)ei[N_EDGES + e]; }
  else             { s = d = e - N_EDGES; }
  float ev = als[s * H + hd] + ald[d * H + hd];
  ev = ev > 0.f ? ev : SLOPE * ev;
  atomicMaxF(&m[d * H + hd], ev);
}

// ---------------- edge pass 2: segment sum of exp ----------------
__global__ void k_edge_sum(const long long* __restrict__ ei, int Hshift, int Hm,
                           const float* __restrict__ als, const float* __restrict__ ald,
                           const float* __restrict__ m, float* __restrict__ sb) {
  int t = blockIdx.x * blockDim.x + threadIdx.x;
  int total = ETOT << Hshift;
  if (t >= total) return;
  int e = t >> Hshift, hd = t & Hm, H = Hm + 1;
  int s, d;
  if (e < N_EDGES) { s = (int)ei[e]; d = (int)ei[N_EDGES + e]; }
  else             { s = d = e - N_EDGES; }
  float ev = als[s * H + hd] + ald[d * H + hd];
  ev = ev > 0.f ? ev : SLOPE * ev;
  atomicAdd(&sb[d * H + hd], __expf(ev - m[d * H + hd]));
}

// ------- edge pass 3: weighted aggregation (1 block per edge, lane=chan) -------
__global__ void k_edge_agg(const long long* __restrict__ ei, int H,
                           const float* __restrict__ als, const float* __restrict__ ald,
                           const float* __restrict__ m, const float* __restrict__ sb,
                           const float* __restrict__ h, float* __restrict__ out) {
  __shared__ int sh_src, sh_dst;
  int e = blockIdx.x;
  if (threadIdx.x == 0) {
    if (e < N_EDGES) { sh_src = (int)ei[e]; sh_dst = (int)ei[N_EDGES + e]; }
    else             { sh_src = sh_dst = e - N_EDGES; }
  }
  __syncthreads();
  int src = sh_src, dst = sh_dst;
  int tid = threadIdx.x;
  int hd  = tid >> 6;                // CC == 64
  int Cw  = blockDim.x;              // H*CC
  float ev = als[src * H + hd] + ald[dst * H + hd];
  ev = ev > 0.f ? ev : SLOPE * ev;
  float alpha = __expf(ev - m[dst * H + hd]) / (sb[dst * H + hd] + 1e-16f);
  atomicAdd(&out[(size_t)dst * Cw + tid], h[(size_t)src * Cw + tid] * alpha);
}

// ---------------- BN stats: partial per-block, lane = channel ----------------
__global__ void k_bn_stats(const float* __restrict__ x, float* __restrict__ sum,
                           float* __restrict__ sumsq, int rows) {
  int c  = threadIdx.x;              // 0..255
  int r0 = blockIdx.x * 256;
  int r1 = r0 + 256 < rows ? r0 + 256 : rows;
  float s1 = 0.f, s2 = 0.f;
  for (int r = r0; r < r1; ++r) {
    float v = x[(size_t)r * HC + c];
    s1 += v; s2 += v * v;
  }
  atomicAdd(&sum[c], s1);
  atomicAdd(&sumsq[c], s2);
}

// -------- fused BN normalize + ELU + optional skip-add (over N*HC) --------
__global__ void k_bn_elu(float* __restrict__ out, const float* __restrict__ in,
                         const float* __restrict__ sum, const float* __restrict__ sumsq,
                         const float* __restrict__ gamma, const float* __restrict__ beta,
                         const float* __restrict__ skip, float invN) {
  int t = blockIdx.x * blockDim.x + threadIdx.x;
  if (t >= N_NODES * HC) return;
  int c = t & (HC - 1);
  float mu  = sum[c] * invN;
  float var = sumsq[c] * invN - mu * mu;
  float v = (in[t] - mu) * rsqrtf(var + EPS_BN) * gamma[c] + beta[c];
  v = v > 0.f ? v : (__expf(v) - 1.f);
  out[t] = v + (skip ? skip[t] : 0.f);
}

// ---------------- final: out = agg3 + b3 + x_skip ----------------
__global__ void k_final(float* __restrict__ out, const float* __restrict__ agg,
                        const float* __restrict__ b3, const float* __restrict__ xskip) {
  int t = blockIdx.x * blockDim.x + threadIdx.x;
  if (t >= N_NODES * CC) return;
  out[t] = agg[t] + b3[t & (CC - 1)] + xskip[t];
}

extern "C" void kernel_launch(void* const* d_in, const int* in_sizes, int n_in,
                              void* d_out, int out_size, void* d_ws, size_t ws_size,
                              hipStream_t stream) {
  const float*     x   = (const float*)d_in[0];
  const long long* ei  = (const long long*)d_in[1];   // int64 [2,E], row-major
  const float* W1  = (const float*)d_in[2];
  const float* as1 = (const float*)d_in[3];
  const float* ad1 = (const float*)d_in[4];
  // d_in[5] = b1  : cancelled exactly by batch_norm mean subtraction
  const float* g1  = (const float*)d_in[6];
  const float* bb1 = (const float*)d_in[7];
  const float* W2  = (const float*)d_in[8];
  const float* as2 = (const float*)d_in[9];
  const float* ad2 = (const float*)d_in[10];
  // d_in[11] = b2 : cancelled by batch_norm
  const float* g2  = (const float*)d_in[12];
  const float* bb2 = (const float*)d_in[13];
  const float* W3  = (const float*)d_in[14];
  const float* as3 = (const float*)d_in[15];
  const float* ad3 = (const float*)d_in[16];
  const float* b3  = (const float*)d_in[17];
  const float* Ws1 = (const float*)d_in[18];
  const float* bs1 = (const float*)d_in[19];
  const float* Ws2 = (const float*)d_in[20];
  const float* bs2 = (const float*)d_in[21];
  float* out = (float*)d_out;

  float* ws = (float*)d_ws;
  const size_t NHC = (size_t)N_NODES * HC;
  float* bufA = ws;                                   // [N,256]
  float* bufB = bufA + NHC;                           // [N,256]
  float* bufC = bufB + NHC;                           // [N,256] (skip storage)
  float* als  = bufC + NHC;                           // [N,4]
  float* ald  = als + (size_t)N_NODES * HD4;          // [N,4]
  float* mb   = ald + (size_t)N_NODES * HD4;          // [N,4]
  float* sb   = mb  + (size_t)N_NODES * HD4;          // [N,4]
  float* bsum = sb  + (size_t)N_NODES * HD4;          // [256]
  float* bssq = bsum + HC;                            // [256]

  const dim3 blk(256);
  const dim3 g256(391, 4), g64(391, 1);               // 3125 row tiles / 8 waves
  const size_t lds128 = (size_t)64 * (IN_CH + PADK) * sizeof(float);  // 33792 B
  const size_t lds256 = (size_t)64 * (HC    + PADK) * sizeof(float);  // 66560 B
  const float invN = 1.0f / (float)N_NODES;
  const float NEG  = -3.0e38f;
  const int nhcI = (int)NHC;

  // ================= layer 1 =================
  k_gemm_wmma<<<g256, blk, lds128, stream>>>(x, Ws1, bs1, bufC, 3125, IN_CH, HC);    // x_init
  k_gemm_wmma<<<g256, blk, lds128, stream>>>(x, W1, nullptr, bufA, 3125, IN_CH, HC); // h1
  k_attn_dots<<<cdiv(N_NODES * HD4 * 32, 256), 256, 0, stream>>>(bufA, as1, ad1, als, ald,
                                                                 N_NODES * HD4, HD4 - 1);
  k_fill<<<cdiv(N_NODES * HD4, 256), 256, 0, stream>>>(mb, NEG, N_NODES * HD4);
  k_fill<<<cdiv(N_NODES * HD4, 256), 256, 0, stream>>>(sb, 0.f, N_NODES * HD4);
  k_fill<<<cdiv(nhcI, 256), 256, 0, stream>>>(bufB, 0.f, nhcI);
  k_edge_max<<<cdiv(ETOT * HD4, 256), 256, 0, stream>>>(ei, 2, 3, als, ald, mb);
  k_edge_sum<<<cdiv(ETOT * HD4, 256), 256, 0, stream>>>(ei, 2, 3, als, ald, mb, sb);
  k_edge_agg<<<ETOT, HC, 0, stream>>>(ei, HD4, als, ald, mb, sb, bufA, bufB);        // agg1
  k_fill<<<1, 256, 0, stream>>>(bsum, 0.f, HC);
  k_fill<<<1, 256, 0, stream>>>(bssq, 0.f, HC);
  k_bn_stats<<<cdiv(N_NODES, 256), 256, 0, stream>>>(bufB, bsum, bssq, N_NODES);
  k_bn_elu<<<cdiv(nhcI, 256), 256, 0, stream>>>(bufA, bufB, bsum, bssq, g1, bb1, bufC, invN);
  // bufA = h (layer-2 input)

  // ================= skip2 + layer 2 =================
  k_gemm_wmma<<<g64, blk, lds256, stream>>>(bufA, Ws2, bs2, bufC, 3125, HC, CC);     // x_skip [N,64]
  k_gemm_wmma<<<g256, blk, lds256, stream>>>(bufA, W2, nullptr, bufB, 3125, HC, HC); // h2
  k_attn_dots<<<cdiv(N_NODES * HD4 * 32, 256), 256, 0, stream>>>(bufB, as2, ad2, als, ald,
                                                                 N_NODES * HD4, HD4 - 1);
  k_fill<<<cdiv(N_NODES * HD4, 256), 256, 0, stream>>>(mb, NEG, N_NODES * HD4);
  k_fill<<<cdiv(N_NODES * HD4, 256), 256, 0, stream>>>(sb, 0.f, N_NODES * HD4);
  k_fill<<<cdiv(nhcI, 256), 256, 0, stream>>>(bufA, 0.f, nhcI);
  k_edge_max<<<cdiv(ETOT * HD4, 256), 256, 0, stream>>>(ei, 2, 3, als, ald, mb);
  k_edge_sum<<<cdiv(ETOT * HD4, 256), 256, 0, stream>>>(ei, 2, 3, als, ald, mb, sb);
  k_edge_agg<<<ETOT, HC, 0, stream>>>(ei, HD4, als, ald, mb, sb, bufB, bufA);        // agg2
  k_fill<<<1, 256, 0, stream>>>(bsum, 0.f, HC);
  k_fill<<<1, 256, 0, stream>>>(bssq, 0.f, HC);
  k_bn_stats<<<cdiv(N_NODES, 256), 256, 0, stream>>>(bufA, bsum, bssq, N_NODES);
  k_bn_elu<<<cdiv(nhcI, 256), 256, 0, stream>>>(bufB, bufA, bsum, bssq, g2, bb2, nullptr, invN);
  // bufB = h (layer-3 input)

  // ================= layer 3 (heads=1) =================
  k_gemm_wmma<<<g64, blk, lds256, stream>>>(bufB, W3, nullptr, bufA, 3125, HC, CC);  // h3 [N,64]
  k_attn_dots<<<cdiv(N_NODES * 32, 256), 256, 0, stream>>>(bufA, as3, ad3, als, ald,
                                                           N_NODES, 0);
  float* agg3 = bufA + (size_t)N_NODES * CC;
  k_fill<<<cdiv(N_NODES, 256), 256, 0, stream>>>(mb, NEG, N_NODES);
  k_fill<<<cdiv(N_NODES, 256), 256, 0, stream>>>(sb, 0.f, N_NODES);
  k_fill<<<cdiv(N_NODES * CC, 256), 256, 0, stream>>>(agg3, 0.f, N_NODES * CC);
  k_edge_max<<<cdiv(ETOT, 256), 256, 0, stream>>>(ei, 0, 0, als, ald, mb);
  k_edge_sum<<<cdiv(ETOT, 256), 256, 0, stream>>>(ei, 0, 0, als, ald, mb, sb);
  k_edge_agg<<<ETOT, CC, 0, stream>>>(ei, 1, als, ald, mb, sb, bufA, agg3);
  k_final<<<cdiv(N_NODES * CC, 256), 256, 0, stream>>>(out, agg3, b3, bufC);
}